// GATConv_69123203662120
// MI455X (gfx1250) — compile-verified
//
#include <hip/hip_runtime.h>
#include <math.h>

typedef float v2f __attribute__((ext_vector_type(2)));
typedef float v8f __attribute__((ext_vector_type(8)));

#define N_NODES   100000
#define N_EDGES   1600000
#define D_IN      128
#define D_OUT     64
#define ROW_BLOCKS (N_NODES / 16)   // 6250, exact

// ---------------------------------------------------------------------------
// Init: zero output accumulator, set m=-inf, denom=0 (must re-init per launch)
// ---------------------------------------------------------------------------
__global__ void k_init(float* __restrict__ out, float* __restrict__ m,
                       float* __restrict__ denom) {
    int i = blockIdx.x * 256 + threadIdx.x;
    if (i < N_NODES * D_OUT) out[i] = 0.0f;
    if (i < N_NODES) { m[i] = -__builtin_inff(); denom[i] = 0.0f; }
}

// ---------------------------------------------------------------------------
// Projection GEMM via V_WMMA_F32_16X16X4_F32.
// One wave handles a 16-row strip of X and all 64 output columns (4 tiles).
// W staged in LDS in k-pair-interleaved layout: element (k,c) lives at
// Wl[(k&~1)*64 + 2*c + (k&1)], so each B fragment {W[kk][c], W[kk+1][c]}
// is one contiguous 8B-aligned v2f -> single ds_load_b64, no fix-up moves.
// ---------------------------------------------------------------------------
__global__ __launch_bounds__(256) void k_gemm(const float* __restrict__ X,
                                              const float* __restrict__ W,
                                              const float* __restrict__ bias,
                                              float* __restrict__ Xp) {
    __shared__ float Wl[D_IN * D_OUT];   // 32 KB
    const int tid = threadIdx.x;
    for (int i = tid; i < D_IN * D_OUT; i += 256) {
        const int k  = i >> 6;          // 0..127
        const int c  = i & 63;          // 0..63
        const int kk = k & ~1;
        Wl[kk * D_OUT + (c << 1) + (k & 1)] = W[i];
    }
    __syncthreads();

    const int wave     = tid >> 5;
    const int lane     = tid & 31;
    const int rowBlock = blockIdx.x * 8 + wave;
    if (rowBlock >= ROW_BLOCKS) return;          // wave-uniform: EXEC stays full

    const int rowBase = rowBlock * 16;
    const int l15     = lane & 15;
    const int khalf   = (lane >> 4) << 1;        // 0 for lanes 0-15, 2 for 16-31

    const float* xrow = X + (size_t)(rowBase + l15) * D_IN;

    v8f acc0 = {}, acc1 = {}, acc2 = {}, acc3 = {};
    for (int k0 = 0; k0 < D_IN; k0 += 4) {
        // A fragment: 16x4 tile, lane holds X[row, k0+khalf .. +1] (8B aligned)
        v2f a = *(const v2f*)(xrow + k0 + khalf);
        // B fragments: one v2f LDS load per column tile (pair-interleaved W)
        const float* wp = Wl + (k0 + khalf) * D_OUT;   // k0+khalf is even
        v2f b0 = *(const v2f*)(wp + (l15 << 1));
        v2f b1 = *(const v2f*)(wp + ((16 + l15) << 1));
        v2f b2 = *(const v2f*)(wp + ((32 + l15) << 1));
        v2f b3 = *(const v2f*)(wp + ((48 + l15) << 1));
        acc0 = __builtin_amdgcn_wmma_f32_16x16x4_f32(false, a, false, b0, (short)0, acc0, false, false);
        acc1 = __builtin_amdgcn_wmma_f32_16x16x4_f32(false, a, false, b1, (short)0, acc1, false, false);
        acc2 = __builtin_amdgcn_wmma_f32_16x16x4_f32(false, a, false, b2, (short)0, acc2, false, false);
        acc3 = __builtin_amdgcn_wmma_f32_16x16x4_f32(false, a, false, b3, (short)0, acc3, false, false);
    }

    // D layout: VGPR r -> lanes 0-15: (M=r, N=lane); lanes 16-31: (M=r+8, N=lane-16)
    const int orow = rowBase + (lane >> 4) * 8;
    const float b0s = bias[l15], b1s = bias[16 + l15];
    const float b2s = bias[32 + l15], b3s = bias[48 + l15];
#pragma unroll
    for (int r = 0; r < 8; ++r) {
        float* o = Xp + (size_t)(orow + r) * D_OUT;
        o[l15]      = acc0[r] + b0s;
        o[16 + l15] = acc1[r] + b1s;
        o[32 + l15] = acc2[r] + b2s;
        o[48 + l15] = acc3[r] + b3s;
    }
}

// ---------------------------------------------------------------------------
// Per-node attention scores: s_src = Xp.a_src + b, s_dst = Xp.a_dst + b.
// One wave per node, lane handles features {lane, lane+32}, xor-shuffle reduce.
// ---------------------------------------------------------------------------
__global__ __launch_bounds__(256) void k_scores(const float* __restrict__ Xp,
                                                const float* __restrict__ a_src,
                                                const float* __restrict__ a_src_b,
                                                const float* __restrict__ a_dst,
                                                const float* __restrict__ a_dst_b,
                                                float* __restrict__ s_src,
                                                float* __restrict__ s_dst) {
    const int lane = threadIdx.x & 31;
    const int node = (blockIdx.x * 256 + threadIdx.x) >> 5;
    if (node >= N_NODES) return;
    const float* row = Xp + (size_t)node * D_OUT;
    const float x0 = row[lane], x1 = row[lane + 32];
    float ps = x0 * a_src[lane] + x1 * a_src[lane + 32];
    float pd = x0 * a_dst[lane] + x1 * a_dst[lane + 32];
#pragma unroll
    for (int off = 16; off > 0; off >>= 1) {
        ps += __shfl_xor(ps, off, 32);
        pd += __shfl_xor(pd, off, 32);
    }
    if (lane == 0) {
        s_src[node] = ps + a_src_b[0];
        s_dst[node] = pd + a_dst_b[0];
    }
}

// ---------------------------------------------------------------------------
// Edge logits: e = leaky_relu(s_src[src]+s_dst[dst]); atomic segment-max into m.
// Order-independent float max via signed int-max / unsigned int-min trick.
// ---------------------------------------------------------------------------
__global__ void k_edge_logits(const int* __restrict__ e_src,
                              const int* __restrict__ e_dst,
                              const float* __restrict__ s_src,
                              const float* __restrict__ s_dst,
                              float* __restrict__ e, float* __restrict__ m) {
    int i = blockIdx.x * 256 + threadIdx.x;
    if (i >= N_EDGES) return;
    float v = s_src[e_src[i]] + s_dst[e_dst[i]];
    v = v > 0.0f ? v : 0.01f * v;
    e[i] = v;
    float* addr = m + e_dst[i];
    if (v >= 0.0f) atomicMax((int*)addr, __float_as_int(v));
    else           atomicMin((unsigned int*)addr, __float_as_uint(v));
}

// ---------------------------------------------------------------------------
// exp(e - m[dst]) and segment-sum of denominators
// ---------------------------------------------------------------------------
__global__ void k_edge_exp(const int* __restrict__ e_dst,
                           float* __restrict__ e,
                           const float* __restrict__ m,
                           float* __restrict__ denom) {
    int i = blockIdx.x * 256 + threadIdx.x;
    if (i >= N_EDGES) return;
    const int d = e_dst[i];
    const float w = expf(e[i] - m[d]);
    e[i] = w;
    atomicAdd(denom + d, w);
}

// ---------------------------------------------------------------------------
// Weighted scatter: out[dst] += (w/denom[dst]) * Xp[src]. One wave per edge,
// lane handles features {lane, lane+32}. Xp/out live in 192MB L2.
// ---------------------------------------------------------------------------
__global__ __launch_bounds__(256) void k_scatter(const int* __restrict__ e_src,
                                                 const int* __restrict__ e_dst,
                                                 const float* __restrict__ e,
                                                 const float* __restrict__ denom,
                                                 const float* __restrict__ Xp,
                                                 float* __restrict__ out) {
    const int lane = threadIdx.x & 31;
    const long long edge = ((long long)blockIdx.x * 256 + threadIdx.x) >> 5;
    if (edge >= N_EDGES) return;
    const int s = e_src[edge], d = e_dst[edge];
    const float coeff = e[edge] / denom[d];
    const float* xs = Xp + (size_t)s * D_OUT;
    float* od = out + (size_t)d * D_OUT;
    atomicAdd(od + lane,      coeff * xs[lane]);
    atomicAdd(od + lane + 32, coeff * xs[lane + 32]);
}

// ---------------------------------------------------------------------------
// Final ELU (in place on d_out)
// ---------------------------------------------------------------------------
__global__ void k_elu(float* __restrict__ out) {
    int i = blockIdx.x * 256 + threadIdx.x;
    if (i >= N_NODES * D_OUT) return;
    const float v = out[i];
    out[i] = v > 0.0f ? v : (expf(v) - 1.0f);
}

// ---------------------------------------------------------------------------
extern "C" void kernel_launch(void* const* d_in, const int* in_sizes, int n_in,
                              void* d_out, int out_size, void* d_ws, size_t ws_size,
                              hipStream_t stream) {
    const float* X       = (const float*)d_in[0];
    const float* theta_w = (const float*)d_in[1];
    const float* theta_b = (const float*)d_in[2];
    const float* a_src   = (const float*)d_in[3];
    const float* a_src_b = (const float*)d_in[4];
    const float* a_dst   = (const float*)d_in[5];
    const float* a_dst_b = (const float*)d_in[6];
    const int*   e_src   = (const int*)d_in[7];
    const int*   e_dst   = (const int*)d_in[8];
    float*       out     = (float*)d_out;

    // Workspace layout (all f32): Xp | s_src | s_dst | m | denom | e
    float* Xp    = (float*)d_ws;
    float* s_src = Xp    + (size_t)N_NODES * D_OUT;
    float* s_dst = s_src + N_NODES;
    float* m     = s_dst + N_NODES;
    float* denom = m     + N_NODES;
    float* e     = denom + N_NODES;

    const int elemGrid  = (N_NODES * D_OUT + 255) / 256;   // 25000
    const int gemmGrid  = (ROW_BLOCKS + 7) / 8;            // 782
    const int nodeWGrid = (N_NODES * 32 + 255) / 256;      // 12500
    const int edgeGrid  = (N_EDGES + 255) / 256;           // 6250
    const int edgeWGrid = (int)(((long long)N_EDGES * 32 + 255) / 256); // 200000

    k_init<<<elemGrid, 256, 0, stream>>>(out, m, denom);
    k_gemm<<<gemmGrid, 256, 0, stream>>>(X, theta_w, theta_b, Xp);
    k_scores<<<nodeWGrid, 256, 0, stream>>>(Xp, a_src, a_src_b, a_dst, a_dst_b, s_src, s_dst);
    k_edge_logits<<<edgeGrid, 256, 0, stream>>>(e_src, e_dst, s_src, s_dst, e, m);
    k_edge_exp<<<edgeGrid, 256, 0, stream>>>(e_dst, e, m, denom);
    k_scatter<<<edgeWGrid, 256, 0, stream>>>(e_src, e_dst, e, denom, Xp, out);
    k_elu<<<elemGrid, 256, 0, stream>>>(out);
}